// FGDLoss_53953379172935
// MI455X (gfx1250) — compile-verified
//
#include <hip/hip_runtime.h>
#include <hip/hip_bf16.h>

typedef __attribute__((ext_vector_type(2))) float v2f;
typedef __attribute__((ext_vector_type(8))) float v8f;

#define BB   8
#define CC   256
#define C2   128
#define HH   160
#define WW   160
#define HWN  25600
#define NBOX 32

// ---- workspace layout (float offsets) ----
#define OFF_FEA_S   0u
#define OFF_FEA_T   204800u
#define OFF_CM_S    409600u
#define OFF_CM_T    614400u
#define OFF_SATT_S  819200u
#define OFF_SATT_T  1024000u
#define OFF_SM_S    1228800u
#define OFF_SM_T    1433600u
#define OFF_AREA    1638400u
#define OFF_BGM     1843200u
#define OFF_CH_S    2048000u
#define OFF_CH_T    2050048u
#define OFF_CAT_S   2052096u
#define OFF_CAT_T   2054144u
#define OFF_CTX_S   2056192u
#define OFF_CTX_T   2058240u
#define OFF_S1      2060288u
#define OFF_S2      2062336u
#define OFF_F       2064384u
#define OFF_BG2     2066432u
#define OFF_MASKSP  2068480u
#define ACC_ZERO_BEGIN OFF_CH_S
#define ACC_ZERO_COUNT (OFF_MASKSP + 1u - OFF_CH_S)
#define HW_ACC_ZERO_COUNT (4u * 204800u)   // fea_s, fea_t, cm_s, cm_t (atomic-accumulated)

__device__ __forceinline__ float wred(float v) {
#pragma unroll
  for (int off = 16; off > 0; off >>= 1) v += __shfl_xor(v, off, 32);
  return v;
}
__device__ __forceinline__ float wredmax(float v) {
#pragma unroll
  for (int off = 16; off > 0; off >>= 1) v = fmaxf(v, __shfl_xor(v, off, 32));
  return v;
}
// 256-thread block reduce; every thread gets the total. sh must have >= 8 floats.
__device__ __forceinline__ float bred(float v, float* sh) {
  __syncthreads();
  int lane = threadIdx.x & 31, wv = threadIdx.x >> 5;
  v = wred(v);
  if (lane == 0) sh[wv] = v;
  __syncthreads();
  float r = sh[lane & 7];
#pragma unroll
  for (int off = 4; off > 0; off >>= 1) r += __shfl_xor(r, off, 32);
  return r;
}
__device__ __forceinline__ float bredmax(float v, float* sh) {
  __syncthreads();
  int lane = threadIdx.x & 31, wv = threadIdx.x >> 5;
  v = wredmax(v);
  if (lane == 0) sh[wv] = v;
  __syncthreads();
  float r = sh[lane & 7];
#pragma unroll
  for (int off = 4; off > 0; off >>= 1) r = fmaxf(r, __shfl_xor(r, off, 32));
  return r;
}

// ---- zero accumulator regions of ws (ws is poisoned, not re-zeroed by harness) ----
__global__ void k_init(float* ws) {
  unsigned stride = gridDim.x * blockDim.x;
  unsigned i = blockIdx.x * blockDim.x + threadIdx.x;
  for (unsigned j = i; j < HW_ACC_ZERO_COUNT; j += stride) ws[j] = 0.f;
  for (unsigned j = i; j < ACC_ZERO_COUNT; j += stride) ws[ACC_ZERO_BEGIN + j] = 0.f;
}

// ---- pass 1: read S,T once (b128); per-hw partials over c + per-(n,c) sums over hw ----
// grid: n(8) x hwchunk(25) x csplit(4) = 800 blocks; lane carries 4 hw positions.
__global__ void __launch_bounds__(256) k_pass1(
    const float* __restrict__ S, const float* __restrict__ T,
    const float* __restrict__ wms, const float* __restrict__ bms,
    const float* __restrict__ wmt, const float* __restrict__ bmt,
    float* __restrict__ ws) {
  __shared__ float lchS[64], lchT[64];
  int tid = threadIdx.x, lane = tid & 31;
  if (tid < 64) { lchS[tid] = 0.f; lchT[tid] = 0.f; }
  __syncthreads();
  int n = blockIdx.x / 100;
  int j = blockIdx.x % 100;
  int hwc = j >> 2, cs = j & 3;
  int c0 = cs << 6;
  int hw = hwc * 1024 + tid * 4;
  size_t base = (size_t)n * CC * HWN + (size_t)c0 * HWN + hw;
  float4 aS = {0,0,0,0}, aT = {0,0,0,0}, cS = {0,0,0,0}, cT = {0,0,0,0};
  for (int c = 0; c < 64; ++c) {
    float4 vs = *reinterpret_cast<const float4*>(S + base + (size_t)c * HWN);
    float4 vt = *reinterpret_cast<const float4*>(T + base + (size_t)c * HWN);
    float fx = fabsf(vs.x), fy = fabsf(vs.y), fz = fabsf(vs.z), fw = fabsf(vs.w);
    float tx = fabsf(vt.x), ty = fabsf(vt.y), tz = fabsf(vt.z), tw = fabsf(vt.w);
    aS.x += fx; aS.y += fy; aS.z += fz; aS.w += fw;
    aT.x += tx; aT.y += ty; aT.z += tz; aT.w += tw;
    float wS = wms[c0 + c], wT = wmt[c0 + c];
    cS.x += vs.x * wS; cS.y += vs.y * wS; cS.z += vs.z * wS; cS.w += vs.w * wS;
    cT.x += vt.x * wT; cT.y += vt.y * wT; cT.z += vt.z * wT; cT.w += vt.w * wT;
    float fsum = (fx + fy) + (fz + fw);
    float tsum = (tx + ty) + (tz + tw);
    fsum = wred(fsum); tsum = wred(tsum);
    if (lane == 0) { atomicAdd(&lchS[c], fsum); atomicAdd(&lchT[c], tsum); }
  }
  __syncthreads();
  if (tid < 64) {
    atomicAdd(&ws[OFF_CH_S + n * CC + c0 + tid], lchS[tid]);
    atomicAdd(&ws[OFF_CH_T + n * CC + c0 + tid], lchT[tid]);
  }
  float bS = (cs == 0) ? bms[0] : 0.f;   // bias added exactly once per element
  float bT = (cs == 0) ? bmt[0] : 0.f;
  size_t p = (size_t)n * HWN + hw;
  atomicAdd(&ws[OFF_FEA_S + p + 0], aS.x * (1.f / CC));
  atomicAdd(&ws[OFF_FEA_S + p + 1], aS.y * (1.f / CC));
  atomicAdd(&ws[OFF_FEA_S + p + 2], aS.z * (1.f / CC));
  atomicAdd(&ws[OFF_FEA_S + p + 3], aS.w * (1.f / CC));
  atomicAdd(&ws[OFF_FEA_T + p + 0], aT.x * (1.f / CC));
  atomicAdd(&ws[OFF_FEA_T + p + 1], aT.y * (1.f / CC));
  atomicAdd(&ws[OFF_FEA_T + p + 2], aT.z * (1.f / CC));
  atomicAdd(&ws[OFF_FEA_T + p + 3], aT.w * (1.f / CC));
  atomicAdd(&ws[OFF_CM_S + p + 0], cS.x + bS);
  atomicAdd(&ws[OFF_CM_S + p + 1], cS.y + bS);
  atomicAdd(&ws[OFF_CM_S + p + 2], cS.z + bS);
  atomicAdd(&ws[OFF_CM_S + p + 3], cS.w + bS);
  atomicAdd(&ws[OFF_CM_T + p + 0], cT.x + bT);
  atomicAdd(&ws[OFF_CM_T + p + 1], cT.y + bT);
  atomicAdd(&ws[OFF_CM_T + p + 2], cT.z + bT);
  atomicAdd(&ws[OFF_CM_T + p + 3], cT.w + bT);
}

// ---- softmax over hw for 4 arrays (s_attn_s/t scaled by HW, sm_s/t) ----
__global__ void __launch_bounds__(256) k_rowsoftmax(float* ws) {
  __shared__ float sh[8];
  int arr = blockIdx.x >> 3, n = blockIdx.x & 7;
  unsigned io, oo; float scale;
  if (arr == 0)      { io = OFF_FEA_S; oo = OFF_SATT_S; scale = (float)HWN; }
  else if (arr == 1) { io = OFF_FEA_T; oo = OFF_SATT_T; scale = (float)HWN; }
  else if (arr == 2) { io = OFF_CM_S;  oo = OFF_SM_S;   scale = 1.f; }
  else               { io = OFF_CM_T;  oo = OFF_SM_T;   scale = 1.f; }
  const float* in = ws + io + (size_t)n * HWN;
  float* out = ws + oo + (size_t)n * HWN;
  int tid = threadIdx.x;
  float m = -1e30f;
  for (int i = tid; i < HWN; i += 256) m = fmaxf(m, in[i]);
  m = bredmax(m, sh);
  float s = 0.f;
  for (int i = tid; i < HWN; i += 256) s += expf(in[i] - m);
  s = bred(s, sh);
  float inv = scale / s;
  for (int i = tid; i < HWN; i += 256) out[i] = expf(in[i] - m) * inv;
}

// ---- channel softmax: c_attn = C * softmax(ch/HW) ----
__global__ void __launch_bounds__(256) k_csoftmax(float* ws) {
  __shared__ float sh[8];
  int which = blockIdx.x >> 3, n = blockIdx.x & 7;
  const float* in = ws + (which ? OFF_CH_T : OFF_CH_S) + n * CC;
  float* out = ws + (which ? OFF_CAT_T : OFF_CAT_S) + n * CC;
  int tid = threadIdx.x;
  float v = in[tid] * (1.0f / HWN);
  float m = bredmax(v, sh);
  float e = expf(v - m);
  float s = bred(e, sh);
  out[tid] = (float)CC * e / s;
}

// ---- bbox masks: area_map (fg) and normalized bg ----
__global__ void __launch_bounds__(256) k_masks(
    const int* __restrict__ bb, const int* __restrict__ ih,
    const int* __restrict__ iw, float* ws) {
  __shared__ int bx1[NBOX], by1[NBOX], bx2[NBOX], by2[NBOX], bval[NBOX];
  __shared__ float bai[NBOX];
  __shared__ float sh[8];
  int n = blockIdx.x, tid = threadIdx.x;
  if (tid < NBOX) {
    float sx = (float)WW / (float)iw[0];
    float sy = (float)HH / (float)ih[0];
    const int* p = bb + (n * NBOX + tid) * 4;
    int x1 = (int)fminf(fmaxf((float)p[0] * sx, 0.f), WW - 1.f);
    int y1 = (int)fminf(fmaxf((float)p[1] * sy, 0.f), HH - 1.f);
    int x2 = (int)fminf(fmaxf((float)p[2] * sx, 0.f), WW - 1.f);
    int y2 = (int)fminf(fmaxf((float)p[3] * sy, 0.f), HH - 1.f);
    bx1[tid] = x1; by1[tid] = y1; bx2[tid] = x2; by2[tid] = y2;
    bval[tid] = (x2 > x1) && (y2 > y1);
    int ar = max(1, y2 - y1 + 1) * max(1, x2 - x1 + 1);
    bai[tid] = 1.f / (float)ar;
  }
  __syncthreads();
  float bgsum = 0.f;
  for (int i = tid; i < HWN; i += 256) {
    int y = i / WW, x = i % WW;
    float amax = 0.f; int fg = 0;
    for (int m = 0; m < NBOX; ++m) {
      bool in = bval[m] && y >= by1[m] && y <= by2[m] && x >= bx1[m] && x <= bx2[m];
      if (in) { fg = 1; amax = fmaxf(amax, bai[m]); }
    }
    ws[OFF_AREA + (size_t)n * HWN + i] = amax;
    float bgv = fg ? 0.f : 1.f;
    ws[OFF_BGM + (size_t)n * HWN + i] = bgv;
    bgsum += bgv;
  }
  bgsum = bred(bgsum, sh);
  float inv = (bgsum > 0.f) ? 1.f / (bgsum + 1e-6f) : 1.f;
  for (int i = tid; i < HWN; i += 256)
    ws[OFF_BGM + (size_t)n * HWN + i] *= inv;
}

// ---- spatial attention L1 diff ----
__global__ void __launch_bounds__(256) k_maskdiff(float* ws) {
  __shared__ float sh[8];
  float a = 0.f;
  for (unsigned i = blockIdx.x * 256u + threadIdx.x; i < BB * HWN; i += gridDim.x * 256u)
    a += fabsf(ws[OFF_SATT_S + i] - ws[OFF_SATT_T + i]);
  a = bred(a, sh);
  if (threadIdx.x == 0) atomicAdd(&ws[OFF_MASKSP], a);
}

// ---- pass 2: read S,T once (b128); per-(n,c): ctx pools + S1,S2,F,B moments ----
__global__ void __launch_bounds__(256) k_pass2(
    const float* __restrict__ S, const float* __restrict__ T,
    float* __restrict__ ws) {
  __shared__ float l[6 * 64];
  int tid = threadIdx.x, lane = tid & 31;
  for (int i = tid; i < 6 * 64; i += 256) l[i] = 0.f;
  __syncthreads();
  int n = blockIdx.x / 100;
  int j = blockIdx.x % 100;
  int hwc = j >> 2, cs = j & 3;
  int c0 = cs << 6;
  int hw = hwc * 1024 + tid * 4;
  size_t p = (size_t)n * HWN + hw;
  float4 smS = *reinterpret_cast<const float4*>(ws + OFF_SM_S + p);
  float4 smT = *reinterpret_cast<const float4*>(ws + OFF_SM_T + p);
  float4 sat = *reinterpret_cast<const float4*>(ws + OFF_SATT_T + p);
  float4 ar  = *reinterpret_cast<const float4*>(ws + OFF_AREA + p);
  float4 bg  = *reinterpret_cast<const float4*>(ws + OFF_BGM + p);
  float4 wfg = {sat.x * ar.x, sat.y * ar.y, sat.z * ar.z, sat.w * ar.w};
  float4 wbg = {sat.x * bg.x, sat.y * bg.y, sat.z * bg.z, sat.w * bg.w};
  size_t base = (size_t)n * CC * HWN + (size_t)c0 * HWN + hw;
  for (int c = 0; c < 64; ++c) {
    float4 vs = *reinterpret_cast<const float4*>(S + base + (size_t)c * HWN);
    float4 vt = *reinterpret_cast<const float4*>(T + base + (size_t)c * HWN);
    float dx = vs.x - vt.x, dy = vs.y - vt.y, dz = vs.z - vt.z, dw = vs.w - vt.w;
    float ex = dx * dx, ey = dy * dy, ez = dz * dz, ew = dw * dw;
    float r0 = (vs.x * smS.x + vs.y * smS.y) + (vs.z * smS.z + vs.w * smS.w);
    float r1 = (vt.x * smT.x + vt.y * smT.y) + (vt.z * smT.z + vt.w * smT.w);
    float r2 = (dx + dy) + (dz + dw);
    float r3 = (ex + ey) + (ez + ew);
    float r4 = (ex * wfg.x + ey * wfg.y) + (ez * wfg.z + ew * wfg.w);
    float r5 = (ex * wbg.x + ey * wbg.y) + (ez * wbg.z + ew * wbg.w);
    r0 = wred(r0); r1 = wred(r1); r2 = wred(r2);
    r3 = wred(r3); r4 = wred(r4); r5 = wred(r5);
    if (lane == 0) {
      atomicAdd(&l[c], r0);
      atomicAdd(&l[64 + c], r1);
      atomicAdd(&l[128 + c], r2);
      atomicAdd(&l[192 + c], r3);
      atomicAdd(&l[256 + c], r4);
      atomicAdd(&l[320 + c], r5);
    }
  }
  __syncthreads();
  if (tid < 64) {
    int o = n * CC + c0 + tid;
    atomicAdd(&ws[OFF_CTX_S + o], l[tid]);
    atomicAdd(&ws[OFF_CTX_T + o], l[64 + tid]);
    atomicAdd(&ws[OFF_S1 + o],    l[128 + tid]);
    atomicAdd(&ws[OFF_S2 + o],    l[192 + tid]);
    atomicAdd(&ws[OFF_F + o],     l[256 + tid]);
    atomicAdd(&ws[OFF_BG2 + o],   l[320 + tid]);
  }
}

// ---- finalize: WMMA fp32 MLPs (ctx -> add) + scalar loss assembly ----
__global__ void __launch_bounds__(256) k_finalize(
    const float* __restrict__ w1s, const float* __restrict__ b1s,
    const float* __restrict__ gs,  const float* __restrict__ bes,
    const float* __restrict__ w2s, const float* __restrict__ b2s,
    const float* __restrict__ w1t, const float* __restrict__ b1t,
    const float* __restrict__ gt,  const float* __restrict__ bet,
    const float* __restrict__ w2t, const float* __restrict__ b2t,
    float* __restrict__ ws, float* __restrict__ out) {
  __shared__ float Xs[16 * CC];    // padded 16x256 input rows
  __shared__ float Ys[16 * C2];    // 16x128 hidden
  __shared__ float addS[BB * CC];
  __shared__ float addT[BB * CC];
  __shared__ float sh[8];
  int tid = threadIdx.x, lane = tid & 31, wv = tid >> 5;
  int hi = lane >> 4, mrow = lane & 15;

  for (int which = 0; which < 2; ++which) {
    const float* ctx = ws + (which ? OFF_CTX_T : OFF_CTX_S);
    const float* w1 = which ? w1t : w1s;
    const float* b1 = which ? b1t : b1s;
    const float* g  = which ? gt  : gs;
    const float* be = which ? bet : bes;
    const float* w2 = which ? w2t : w2s;
    const float* b2 = which ? b2t : b2s;
    float* outAdd = which ? addT : addS;

    for (int i = tid; i < 16 * CC; i += 256) {
      int r = i >> 8;
      Xs[i] = (r < BB) ? ctx[(r << 8) + (i & 255)] : 0.f;
    }
    __syncthreads();

    // GEMM1: Y(16x128) = X(16x256) @ W1^T + b1 ; wave wv -> cols [16wv,16wv+16)
    {
      int col = wv * 16 + mrow;
      v8f acc = {0.f, 0.f, 0.f, 0.f, 0.f, 0.f, 0.f, 0.f};
      for (int k = 0; k < CC; k += 4) {
        int kk = k + 2 * hi;
        v2f a, b;
        a.x = Xs[mrow * CC + kk];
        a.y = Xs[mrow * CC + kk + 1];
        b.x = w1[col * CC + kk];
        b.y = w1[col * CC + kk + 1];
        acc = __builtin_amdgcn_wmma_f32_16x16x4_f32(
            false, a, false, b, (short)0, acc, false, false);
      }
      float bias = b1[col];
#pragma unroll
      for (int r = 0; r < 8; ++r)
        Ys[(r + 8 * hi) * C2 + col] = acc[r] + bias;
    }
    __syncthreads();

    // LayerNorm + ReLU per row (over 128 cols); wave wv does rows wv, wv+8
    for (int rr = wv; rr < 16; rr += 8) {
      float s = 0.f, s2 = 0.f;
      float vals[4];
#pragma unroll
      for (int j = 0; j < 4; ++j) {
        float y = Ys[rr * C2 + lane + 32 * j];
        vals[j] = y; s += y; s2 += y * y;
      }
      s = wred(s); s2 = wred(s2);
      float mu = s * (1.f / C2);
      float var = fmaxf(s2 * (1.f / C2) - mu * mu, 0.f);
      float rs = rsqrtf(var + 1e-5f);
#pragma unroll
      for (int j = 0; j < 4; ++j) {
        int cidx = lane + 32 * j;
        float z = (vals[j] - mu) * rs * g[cidx] + be[cidx];
        Ys[rr * C2 + cidx] = fmaxf(z, 0.f);
      }
    }
    __syncthreads();

    // GEMM2: Out(16x256) = Z(16x128) @ W2^T + b2 ; wave wv -> tiles wv, wv+8
    for (int t = wv; t < 16; t += 8) {
      int col = t * 16 + mrow;
      v8f acc = {0.f, 0.f, 0.f, 0.f, 0.f, 0.f, 0.f, 0.f};
      for (int k = 0; k < C2; k += 4) {
        int kk = k + 2 * hi;
        v2f a, b;
        a.x = Ys[mrow * C2 + kk];
        a.y = Ys[mrow * C2 + kk + 1];
        b.x = w2[col * C2 + kk];
        b.y = w2[col * C2 + kk + 1];
        acc = __builtin_amdgcn_wmma_f32_16x16x4_f32(
            false, a, false, b, (short)0, acc, false, false);
      }
      if (hi == 0) {               // rows 0..7 are the real batch rows
        float bias = b2[col];
#pragma unroll
        for (int r = 0; r < 8; ++r)
          outAdd[r * CC + col] = acc[r] + bias;
      }
    }
    __syncthreads();
  }

  // loss = [ a*Σ cat*F + b*Σ cat*B + g*(Σ|cdiff| + mask_sp)
  //          + l*Σ (S2 + 2eS1 + HW e^2) ] / B
  float accL = 0.f;
  for (int i = tid; i < BB * CC; i += 256) {
    float e   = addS[i] - addT[i];
    float s1  = ws[OFF_S1 + i],  s2v = ws[OFF_S2 + i];
    float fv  = ws[OFF_F + i],   bgv = ws[OFF_BG2 + i];
    float cat = ws[OFF_CAT_T + i], cas = ws[OFF_CAT_S + i];
    accL += 0.5f * (s2v + 2.f * e * s1 + (float)HWN * e * e); // LAMBDA_RELA
    accL += cat * fv + 0.25f * cat * bgv;                     // ALPHA, BETA
    accL += 0.5f * fabsf(cas - cat);                          // GAMMA (chan)
  }
  accL = bred(accL, sh);
  if (tid == 0)
    out[0] = (accL + 0.5f * ws[OFF_MASKSP]) * (1.f / BB);
}

extern "C" void kernel_launch(void* const* d_in, const int* in_sizes, int n_in,
                              void* d_out, int out_size, void* d_ws, size_t ws_size,
                              hipStream_t stream) {
  const float* S   = (const float*)d_in[0];
  const float* T   = (const float*)d_in[1];
  const int* bbx   = (const int*)d_in[2];
  const float* wms = (const float*)d_in[3];
  const float* bms = (const float*)d_in[4];
  const float* wmt = (const float*)d_in[5];
  const float* bmt = (const float*)d_in[6];
  const float* w1s = (const float*)d_in[7];
  const float* b1s = (const float*)d_in[8];
  const float* gs  = (const float*)d_in[9];
  const float* bes = (const float*)d_in[10];
  const float* w2s = (const float*)d_in[11];
  const float* b2s = (const float*)d_in[12];
  const float* w1t = (const float*)d_in[13];
  const float* b1t = (const float*)d_in[14];
  const float* gt  = (const float*)d_in[15];
  const float* bet = (const float*)d_in[16];
  const float* w2t = (const float*)d_in[17];
  const float* b2t = (const float*)d_in[18];
  const int* ih    = (const int*)d_in[19];
  const int* iw    = (const int*)d_in[20];
  float* ws  = (float*)d_ws;
  float* out = (float*)d_out;

  k_init<<<512, 256, 0, stream>>>(ws);
  k_pass1<<<800, 256, 0, stream>>>(S, T, wms, bms, wmt, bmt, ws);
  k_rowsoftmax<<<32, 256, 0, stream>>>(ws);
  k_csoftmax<<<16, 256, 0, stream>>>(ws);
  k_masks<<<BB, 256, 0, stream>>>(bbx, ih, iw, ws);
  k_maskdiff<<<64, 256, 0, stream>>>(ws);
  k_pass2<<<800, 256, 0, stream>>>(S, T, ws);
  k_finalize<<<1, 256, 0, stream>>>(w1s, b1s, gs, bes, w2s, b2s,
                                    w1t, b1t, gt, bet, w2t, b2t, ws, out);
}